// PatchEmbedding_25383256719624
// MI455X (gfx1250) — compile-verified
//
#include <hip/hip_runtime.h>

typedef __attribute__((ext_vector_type(2))) float v2f;
typedef __attribute__((ext_vector_type(8))) float v8f;

#define EMB   768
#define IMG   384
#define L1V   383
#define LTOT  146689   // 383*383
#define LTILES 9169    // ceil(LTOT/16)
#define ETILES 48      // 768/16
#define NTILES (LTILES * ETILES)

__global__ __launch_bounds__(256) void patch_embed_wmma(
    const float* __restrict__ x,     // (3,384,384)
    const float* __restrict__ W,     // (768,12)
    const float* __restrict__ bias,  // (768,)
    const float* __restrict__ pos,   // (768,146689)
    float* __restrict__ out)         // (768,146689)
{
    const int lane = threadIdx.x & 31;
    const int wib  = threadIdx.x >> 5;
    const int nwav = (gridDim.x * blockDim.x) >> 5;
    const int wid  = blockIdx.x * (blockDim.x >> 5) + wib;

    const int n = lane & 15;   // tile column (N) / A-matrix row (M)
    const int g = lane >> 4;   // lane group (selects K pair / M+8)

    for (int t = wid; t < NTILES; t += nwav) {
        const int lt = t / ETILES;
        const int et = t - lt * ETILES;
        const int e0 = et << 4;
        const int l0 = lt << 4;

        const int  l     = l0 + n;
        const bool valid = (l < LTOT);
        const unsigned lc = valid ? (unsigned)l : (unsigned)(LTOT - 1);
        const unsigned r  = lc / L1V;            // patch row
        const unsigned c  = lc - r * L1V;        // patch col

        // ---- A tiles: W 16x4 per channel; lane (m=n, g): VGPR v holds K = 2g+v ----
        const float* Wr = W + (size_t)(e0 + n) * 12 + 2 * g;
        v2f a0, a1, a2;
        a0.x = Wr[0]; a0.y = Wr[1];
        a1.x = Wr[4]; a1.y = Wr[5];
        a2.x = Wr[8]; a2.y = Wr[9];

        // ---- B tiles: 4x16 patch data; entry (K=2g+v, N=n) = x[ch, r+g, c+v] ----
        const float* xp = x + (size_t)(r + g) * IMG + c;
        v2f b0, b1, b2;
        b0.x = xp[0];               b0.y = xp[1];
        b1.x = xp[IMG * IMG];       b1.y = xp[IMG * IMG + 1];
        b2.x = xp[2 * IMG * IMG];   b2.y = xp[2 * IMG * IMG + 1];

        // ---- C init = bias broadcast across N; lane's row M = 8g + v ----
        v8f acc;
        const float4 q0 = *reinterpret_cast<const float4*>(bias + e0 + 8 * g);
        const float4 q1 = *reinterpret_cast<const float4*>(bias + e0 + 8 * g + 4);
        acc[0] = q0.x; acc[1] = q0.y; acc[2] = q0.z; acc[3] = q0.w;
        acc[4] = q1.x; acc[5] = q1.y; acc[6] = q1.z; acc[7] = q1.w;

        // ---- D = A0*B0 + A1*B1 + A2*B2 + C   (K = 12 over three K=4 fp32 WMMAs) ----
        acc = __builtin_amdgcn_wmma_f32_16x16x4_f32(false, a0, false, b0, (short)0, acc, false, false);
        acc = __builtin_amdgcn_wmma_f32_16x16x4_f32(false, a1, false, b1, (short)0, acc, false, false);
        acc = __builtin_amdgcn_wmma_f32_16x16x4_f32(false, a2, false, b2, (short)0, acc, false, false);

        // ---- epilogue: += positions (streaming NT), store (streaming NT) ----
        const size_t base = (size_t)(e0 + 8 * g) * LTOT + lc;
#pragma unroll
        for (int v = 0; v < 8; ++v) {
            const size_t idx = base + (size_t)v * LTOT;
            const float val = acc[v] + __builtin_nontemporal_load(pos + idx);
            if (valid) __builtin_nontemporal_store(val, out + idx);
        }
    }
}

extern "C" void kernel_launch(void* const* d_in, const int* in_sizes, int n_in,
                              void* d_out, int out_size, void* d_ws, size_t ws_size,
                              hipStream_t stream) {
    (void)in_sizes; (void)n_in; (void)d_ws; (void)ws_size; (void)out_size;
    const float* x   = (const float*)d_in[0];
    const float* W   = (const float*)d_in[1];
    const float* b   = (const float*)d_in[2];
    const float* pos = (const float*)d_in[3];
    float* out = (float*)d_out;

    dim3 grid(4096);
    dim3 block(256);
    hipLaunchKernelGGL(patch_embed_wmma, grid, block, 0, stream, x, W, b, pos, out);
}